// HashEncoderTaichiNerf_54099408060466
// MI455X (gfx1250) — compile-verified
//
#include <hip/hip_runtime.h>
#include <math.h>

// ---------------------------------------------------------------------------
// Instant-NGP multires hash encoding for MI455X (gfx1250).
//  Roofline: 16 levels x 8 x 8B gathers/query = 1 GB of gathers against a
//  64 MB table set that is fully L2-resident (192 MB L2). HBM traffic is only
//  ~204 MB (x + table fill + out) -> ~9us floor; the limiter is L2 gather
//  throughput, so: fully unrolled straight-line level code (max outstanding
//  global_load_b64), NT stores for the 128 MB output (don't evict tables),
//  TDM async staging of coordinates, LDS transpose for coalesced b128 stores.
//  WMMA is inapplicable: the per-point 8x2 contraction has per-lane data on
//  both operands (no wave-shared matrix).
// ---------------------------------------------------------------------------

#define NUM_LEVELS 16
#define DENSE_LEVELS 7          // levels 0..6 dense (<= 65^3 cells), 7..15 hashed
                                // (>= 81^3 > 2^19); robust to +-1 res rounding
#define TABLE_SIZE (1u << 19)
#define TABLE_MASK (TABLE_SIZE - 1u)
#define BLOCK 256
#define P1 2654435761u
#define P2 805459861u

typedef __attribute__((ext_vector_type(4))) float        v4f;
typedef __attribute__((ext_vector_type(4))) unsigned int v4u;
typedef __attribute__((ext_vector_type(8))) int          v8i;
typedef __attribute__((ext_vector_type(4))) int          v4i;

struct HgParams {
  int res[NUM_LEVELS];
};

__global__ __launch_bounds__(BLOCK) void hashgrid_encode_kernel(
    const float* __restrict__ x, const float* __restrict__ table,
    float* __restrict__ out, int N, HgParams p)
{
  __shared__ float s_x[BLOCK * 3];        // staged coordinates (3 KB)
  __shared__ float s_out[BLOCK * 33];     // 33-float padded rows -> conflict-free

  const int tid  = threadIdx.x;
  const int base = blockIdx.x * BLOCK;

  // ---- Stage x[base*3 .. base*3+768) into LDS via the Tensor Data Mover ----
#if __has_builtin(__builtin_amdgcn_tensor_load_to_lds) && __has_builtin(__builtin_amdgcn_s_wait_tensorcnt)
  if (tid < 32) {  // wave 0 issues the DMA (TDM ignores EXEC; per-wave issue)
    unsigned long long gaddr = (unsigned long long)(const void*)(x + (size_t)base * 3);
    unsigned lds_addr = (unsigned)(unsigned long long)(void*)&s_x[0];
    unsigned rem = (unsigned)((size_t)N * 3 - (size_t)base * 3);  // elements left (>0)

    // D# group 0: count=1, lds_addr, global_addr[56:0], type=2 ("image")
    v4u g0;
    g0[0] = 1u;
    g0[1] = lds_addr;
    g0[2] = (unsigned)(gaddr & 0xFFFFFFFFull);
    g0[3] = (unsigned)((gaddr >> 32) & 0x01FFFFFFull) | (2u << 30);

    // D# group 1: mask=0, data_size=4B, tensor_dim0=rem (HW zero-fills OOB
    // for the tail block), tensor_dim1=1, tile_dim0=768, tile_dim1=1,
    // tensor_dim0_stride=768.
    const unsigned tile0 = (unsigned)(BLOCK * 3);
    v8i g1;
    g1[0] = (int)(2u << 16);
    g1[1] = (int)((rem & 0xFFFFu) << 16);
    g1[2] = (int)((rem >> 16) | (1u << 16));
    g1[3] = (int)(tile0 << 16);
    g1[4] = (int)1u;
    g1[5] = (int)tile0;
    g1[6] = 0;
    g1[7] = 0;

    // D# groups 2/3: higher dims unused (dims=1, tiles=0, strides=0)
    v4i g2; g2[0] = 1; g2[1] = 1; g2[2] = 0; g2[3] = 0;
    v4i g3; g3[0] = 0; g3[1] = (int)(1u << 16); g3[2] = 0; g3[3] = 0;

    // 6-arg toolchain form: trailing int32x8 block only used for
    // multi-descriptor chains; count=1 in group0 -> zero-fill it.
    v8i gx; gx[0] = 0; gx[1] = 0; gx[2] = 0; gx[3] = 0;
    gx[4] = 0; gx[5] = 0; gx[6] = 0; gx[7] = 0;

    __builtin_amdgcn_tensor_load_to_lds(g0, g1, g2, g3, gx, 0);
    __builtin_amdgcn_s_wait_tensorcnt(0);
  }
#else
  for (int j = tid; j < BLOCK * 3; j += BLOCK) {
    size_t g = (size_t)base * 3 + j;
    s_x[j] = (g < (size_t)N * 3) ? x[g] : 0.f;
  }
#endif
  __syncthreads();

  // ---- Per-query encode: 16 levels x 8 corner gathers, straight-line ----
  const int n = base + tid;
  if (n < N) {
    const float xx = s_x[tid * 3 + 0];
    const float xy = s_x[tid * 3 + 1];
    const float xz = s_x[tid * 3 + 2];

#pragma unroll
    for (int l = 0; l < NUM_LEVELS; ++l) {
      const int   res = p.res[l];
      const float rf  = (float)res;                 // matches res.astype(f32)
      const float sx = xx * rf, sy = xy * rf, sz = xz * rf;
      const float fx = floorf(sx), fy = floorf(sy), fz = floorf(sz);
      const float wx = sx - fx, wy = sy - fy, wz = sz - fz;
      const unsigned px = (unsigned)fx, py = (unsigned)fy, pz = (unsigned)fz;

      const float2* __restrict__ tab = (const float2*)table + (size_t)l * TABLE_SIZE;

      // Hoisted per-axis index terms: corner index = tx ^/+ ty ^/+ tz.
      unsigned ty0, ty1, tz0, tz1, tx0, tx1;
      if (l < DENSE_LEVELS) {                       // compile-time after unroll
        const unsigned sY = (unsigned)res + 1u;
        const unsigned sZ = sY * sY;
        tx0 = px;        tx1 = px + 1u;
        ty0 = py * sY;   ty1 = ty0 + sY;
        tz0 = pz * sZ;   tz1 = tz0 + sZ;
      } else {
        tx0 = px;        tx1 = px + 1u;
        ty0 = py * P1;   ty1 = ty0 + P1;
        tz0 = pz * P2;   tz1 = tz0 + P2;
      }

      // Factored trilinear weights; association (selx*sely)*selz is exactly
      // the reference's jnp.prod reduction order -> bit-identical weights.
      const float w0x = 1.f - wx, w0y = 1.f - wy, w0z = 1.f - wz;
      const float a00 = w0x * w0y, a01 = w0x * wy;
      const float a10 = wx * w0y,  a11 = wx * wy;
      float f0 = 0.f, f1 = 0.f;

#pragma unroll
      for (int k = 0; k < 8; ++k) {
        const unsigned bx = (k >> 2) & 1u, by = (k >> 1) & 1u, bz = k & 1u;
        const unsigned tx = bx ? tx1 : tx0;
        const unsigned ty = by ? ty1 : ty0;
        const unsigned tz = bz ? tz1 : tz0;
        const unsigned idx = (l < DENSE_LEVELS) ? (tx + ty + tz)
                                                : ((tx ^ ty ^ tz) & TABLE_MASK);
        const float2 f = tab[idx];                  // global_load_b64 gather
        const float axy = by ? (bx ? a11 : a01) : (bx ? a10 : a00);
        const float w   = axy * (bz ? wz : w0z);
        f0 = fmaf(w, f.x, f0);
        f1 = fmaf(w, f.y, f1);
      }
      s_out[tid * 33 + 2 * l + 0] = f0;             // conflict-free (stride 33)
      s_out[tid * 33 + 2 * l + 1] = f1;
    }
  }
  __syncthreads();

  // ---- Coalesced nontemporal b128 stores of the block's [m,32] tile ----
  const int m = min(BLOCK, N - base);
  if (m > 0) {
    v4f* __restrict__ out4 = (v4f*)out + (size_t)base * 8;
    const int tot = m * 8;
    for (int j = tid; j < tot; j += BLOCK) {
      const int q = j >> 3;
      const int c = (j & 7) << 2;
      const float* row = &s_out[q * 33 + c];
      v4f v = { row[0], row[1], row[2], row[3] };
      __builtin_nontemporal_store(v, out4 + j);     // keep tables L2-resident
    }
  }
}

extern "C" void kernel_launch(void* const* d_in, const int* in_sizes, int n_in,
                              void* d_out, int out_size, void* d_ws, size_t ws_size,
                              hipStream_t stream) {
  const float* x     = (const float*)d_in[0];
  const float* table = (const float*)d_in[1];
  float* out         = (float*)d_out;
  const int N = in_sizes[0] / 3;

  // Per-level resolutions computed with the SAME double-precision
  // exp/log/pow/floor pipeline numpy uses, so boundary levels (exact powers
  // of two) round identically to the reference. The dense/hash split is
  // compile-time (levels 0..6 dense): dense levels have <= 65^3 = 274625
  // cells and the first hashed level has >= 81^3 = 531441 > 2^19, margins
  // far beyond any +-1 ulp rounding of the resolutions.
  HgParams p;
  const double b = exp((log(512.0) - log(16.0)) / 15.0);
  for (int l = 0; l < NUM_LEVELS; ++l) {
    p.res[l] = (int)(long long)floor(16.0 * pow(b, (double)l));
  }

  const int blocks = (N + BLOCK - 1) / BLOCK;
  hipLaunchKernelGGL(hashgrid_encode_kernel, dim3(blocks), dim3(BLOCK), 0, stream,
                     x, table, out, N, p);
}